// GAT_89627377533180
// MI455X (gfx1250) — compile-verified
//
#include <hip/hip_runtime.h>
#include <hip/hip_bf16.h>
#include <stdint.h>

// ---------------- problem constants (from reference) ----------------
constexpr int NN   = 50000;        // nodes
constexpr int EE   = 800000;       // edges (before self loops)
constexpr int GG   = 64;           // graphs
constexpr int HH   = 4;            // heads
constexpr int DD   = 128;          // feature dim (== H*OUT)
constexpr int OUTC = 32;           // per-head channels
constexpr int LL   = 3;            // layers
constexpr int ET   = EE + NN;      // edges incl. self loops
constexpr float NEG_SLOPE = 0.2f;

typedef float v2f __attribute__((ext_vector_type(2)));
typedef float v8f __attribute__((ext_vector_type(8)));

// ---------------- helpers ----------------
__device__ inline void atomicMaxF(float* addr, float val) {
    // monotone ordered-int transform; valid for finite floats
    int iv = __float_as_int(val);
    if (iv >= 0) atomicMax((int*)addr, iv);
    else         atomicMin((unsigned int*)addr, (unsigned int)iv);
}

__global__ __launch_bounds__(256) void k_fill(float* __restrict__ p, float v, int n) {
    int i = blockIdx.x * blockDim.x + threadIdx.x;
    if (i < n) p[i] = v;
}

// ---------------- GEMM: H = X[NN,128] @ W[128,128] via WMMA fp32 ----------------
// block = 256 threads = 8 waves. Block computes rows [row0,row0+16) x all 128 cols.
// Wave w computes the 16x16 tile at columns [16w, 16w+16).
__global__ __launch_bounds__(256) void k_gemm(const float* __restrict__ X,
                                              const float* __restrict__ W,
                                              float* __restrict__ Hm) {
    __shared__ float sW[64 * 128];   // K-chunk x N   (32 KB)
    __shared__ float sX[16 * 64];    // M x K-chunk   (4 KB)
    const int tid  = threadIdx.x;
    const int wave = tid >> 5;
    const int lane = tid & 31;
    const int row0 = blockIdx.x * 16;

    v8f acc = {};
    for (int kc = 0; kc < DD; kc += 64) {
        __syncthreads();
        for (int i = tid; i < 64 * 128; i += 256) {
            int r = i >> 7, c = i & 127;
            sW[i] = W[(kc + r) * DD + c];
        }
        for (int i = tid; i < 16 * 64; i += 256) {
            int r = i >> 6, c = i & 63;
            int gr = row0 + r;
            sX[i] = (gr < NN) ? X[gr * DD + kc + c] : 0.0f;
        }
        __syncthreads();

        const int n0 = wave * 16;
        #pragma unroll
        for (int k4 = 0; k4 < 64; k4 += 4) {
            // A 16x4 fp32 (ISA 7.12.2): lanes 0-15 -> M=lane, K={k4,k4+1};
            //                           lanes 16-31 -> M=lane-16, K={k4+2,k4+3}
            int ar   = lane & 15;
            int koff = k4 + ((lane >> 4) << 1);
            v2f a;
            a.x = sX[ar * 64 + koff];
            a.y = sX[ar * 64 + koff + 1];
            // B 4x16 fp32 (mirrored): lanes 0-15 -> N=n0+lane, VGPR r holds K=koff+r
            int bn = n0 + (lane & 15);
            v2f b;
            b.x = sW[koff * 128 + bn];
            b.y = sW[(koff + 1) * 128 + bn];
            acc = __builtin_amdgcn_wmma_f32_16x16x4_f32(
                false, a, false, b, (short)0, acc, false, false);
        }
    }
    // C/D layout: VGPR r -> lanes 0-15: M=r, N=lane; lanes 16-31: M=r+8, N=lane-16
    const int cn = wave * 16 + (lane & 15);
    const int mb = (lane >> 4) * 8;
    #pragma unroll
    for (int r = 0; r < 8; ++r) {
        int gr = row0 + mb + r;
        if (gr < NN) Hm[gr * DD + cn] = acc[r];
    }
}

// ---------------- per-(node,head) attention halves ----------------
__global__ __launch_bounds__(256) void k_logits(const float* __restrict__ Hm,
                                                const float* __restrict__ a_s,
                                                const float* __restrict__ a_d,
                                                float* __restrict__ als,
                                                float* __restrict__ ald) {
    int idx = blockIdx.x * blockDim.x + threadIdx.x;
    if (idx >= NN * HH) return;
    int n = idx / HH, h = idx % HH;
    const float* hv = Hm + n * DD + h * OUTC;
    float s1 = 0.f, s2 = 0.f;
    #pragma unroll
    for (int i = 0; i < OUTC; ++i) {
        float v = hv[i];
        s1 += v * a_s[h * OUTC + i];
        s2 += v * a_d[h * OUTC + i];
    }
    als[idx] = s1;
    ald[idx] = s2;
}

__device__ inline void edge_nodes(int e, const int* __restrict__ src,
                                  const int* __restrict__ dst, int& sn, int& dn) {
    if (e < EE) { sn = src[e]; dn = dst[e]; }
    else        { sn = e - EE; dn = e - EE; }   // self loop
}

__device__ inline float edge_logit(int sn, int dn, int h,
                                   const float* __restrict__ als,
                                   const float* __restrict__ ald) {
    float lg = als[sn * HH + h] + ald[dn * HH + h];
    return lg > 0.f ? lg : NEG_SLOPE * lg;
}

// ---------------- edge pass 1: segment max ----------------
__global__ __launch_bounds__(256) void k_edge_max(const int* __restrict__ src,
                                                  const int* __restrict__ dst,
                                                  const float* __restrict__ als,
                                                  const float* __restrict__ ald,
                                                  float* __restrict__ m) {
    int idx = blockIdx.x * blockDim.x + threadIdx.x;
    if (idx >= ET * HH) return;
    int e = idx / HH, h = idx % HH;
    int sn, dn; edge_nodes(e, src, dst, sn, dn);
    atomicMaxF(&m[dn * HH + h], edge_logit(sn, dn, h, als, ald));
}

// ---------------- edge pass 2: segment sum of exp ----------------
__global__ __launch_bounds__(256) void k_edge_sum(const int* __restrict__ src,
                                                  const int* __restrict__ dst,
                                                  const float* __restrict__ als,
                                                  const float* __restrict__ ald,
                                                  const float* __restrict__ m,
                                                  float* __restrict__ s) {
    int idx = blockIdx.x * blockDim.x + threadIdx.x;
    if (idx >= ET * HH) return;
    int e = idx / HH, h = idx % HH;
    int sn, dn; edge_nodes(e, src, dst, sn, dn);
    float p = expf(edge_logit(sn, dn, h, als, ald) - m[dn * HH + h]);
    atomicAdd(&s[dn * HH + h], p);
}

// ---------------- edge pass 3: out[dst] += alpha * h[src] ----------------
__global__ __launch_bounds__(256) void k_edge_agg(const int* __restrict__ src,
                                                  const int* __restrict__ dst,
                                                  const float* __restrict__ als,
                                                  const float* __restrict__ ald,
                                                  const float* __restrict__ m,
                                                  const float* __restrict__ s,
                                                  const float* __restrict__ Hm,
                                                  float* __restrict__ acc) {
    int idx = blockIdx.x * blockDim.x + threadIdx.x;
    if (idx >= ET * DD) return;
    int e = idx / DD, c = idx % DD;
    int h = c >> 5;   // c / OUTC
    int sn, dn; edge_nodes(e, src, dst, sn, dn);
    float p = expf(edge_logit(sn, dn, h, als, ald) - m[dn * HH + h]);
    float alpha = p / s[dn * HH + h];
    atomicAdd(&acc[dn * DD + c], alpha * Hm[sn * DD + c]);
}

// ---------------- finalize: relu(acc + bias) -> dst ----------------
__global__ __launch_bounds__(256) void k_finalize(const float* __restrict__ acc,
                                                  const float* __restrict__ bias,
                                                  float* __restrict__ dstp) {
    int i = blockIdx.x * blockDim.x + threadIdx.x;
    if (i >= NN * DD) return;
    float v = acc[i] + bias[i & (DD - 1)];
    dstp[i] = v > 0.f ? v : 0.f;
}

// ---------------- graph pooling ----------------
__global__ __launch_bounds__(256) void k_count(const int* __restrict__ batch,
                                               float* __restrict__ gcnt) {
    int i = blockIdx.x * blockDim.x + threadIdx.x;
    if (i < NN) atomicAdd(&gcnt[batch[i]], 1.0f);
}

__global__ __launch_bounds__(256) void k_gsum(const float* __restrict__ node,
                                              const int* __restrict__ batch,
                                              float* __restrict__ gsum) {
    int i = blockIdx.x * blockDim.x + threadIdx.x;
    if (i >= NN * DD) return;
    int n = i / DD, c = i % DD;
    atomicAdd(&gsum[batch[n] * DD + c], node[i]);
}

__global__ __launch_bounds__(256) void k_gdiv(const float* __restrict__ gsum,
                                              const float* __restrict__ gcnt,
                                              float* __restrict__ outg) {
    int i = blockIdx.x * blockDim.x + threadIdx.x;
    if (i >= GG * DD) return;
    outg[i] = gsum[i] / fmaxf(gcnt[i / DD], 1.0f);
}

// ---------------- launcher ----------------
static inline int cdiv(long long a, int b) { return (int)((a + b - 1) / b); }

extern "C" void kernel_launch(void* const* d_in, const int* in_sizes, int n_in,
                              void* d_out, int out_size, void* d_ws, size_t ws_size,
                              hipStream_t stream) {
    const float* x0    = (const float*)d_in[0];   // [NN,128]
    const int*   ei    = (const int*)d_in[1];     // [2,EE]
    const int*   batch = (const int*)d_in[2];     // [NN]
    const float* Ws    = (const float*)d_in[3];   // [LL,128,128]
    const float* a_src = (const float*)d_in[4];   // [LL,HH,OUTC]
    const float* a_dst = (const float*)d_in[5];   // [LL,HH,OUTC]
    const float* bias  = (const float*)d_in[6];   // [LL,128]
    float* out = (float*)d_out;                   // [NN*128] ++ [GG*128]

    const int* src = ei;
    const int* dst = ei + EE;

    float* ws   = (float*)d_ws;
    float* bufH = ws;                         // NN*DD   (h = xW)
    float* bufA = bufH + (size_t)NN * DD;     // NN*DD   (accumulator / next-x)
    float* als  = bufA + (size_t)NN * DD;     // NN*HH
    float* ald  = als  + (size_t)NN * HH;     // NN*HH
    float* mbuf = ald  + (size_t)NN * HH;     // NN*HH
    float* sbuf = mbuf + (size_t)NN * HH;     // NN*HH
    float* gsum = sbuf + (size_t)NN * HH;     // GG*DD
    float* gcnt = gsum + (size_t)GG * DD;     // GG

    for (int l = 0; l < LL; ++l) {
        const float* xin = (l == 0) ? x0 : bufA;
        k_gemm<<<cdiv(NN, 16), 256, 0, stream>>>(xin, Ws + (size_t)l * DD * DD, bufH);
        k_fill<<<cdiv((long long)NN * DD, 256), 256, 0, stream>>>(bufA, 0.0f, NN * DD);
        k_fill<<<cdiv(NN * HH, 256), 256, 0, stream>>>(mbuf, -1e30f, NN * HH);
        k_fill<<<cdiv(NN * HH, 256), 256, 0, stream>>>(sbuf, 0.0f, NN * HH);
        k_logits<<<cdiv(NN * HH, 256), 256, 0, stream>>>(
            bufH, a_src + (size_t)l * HH * OUTC, a_dst + (size_t)l * HH * OUTC, als, ald);
        k_edge_max<<<cdiv((long long)ET * HH, 256), 256, 0, stream>>>(src, dst, als, ald, mbuf);
        k_edge_sum<<<cdiv((long long)ET * HH, 256), 256, 0, stream>>>(src, dst, als, ald, mbuf, sbuf);
        k_edge_agg<<<cdiv((long long)ET * DD, 256), 256, 0, stream>>>(
            src, dst, als, ald, mbuf, sbuf, bufH, bufA);
        float* fdst = (l == LL - 1) ? out : bufA;
        k_finalize<<<cdiv((long long)NN * DD, 256), 256, 0, stream>>>(bufA, bias + (size_t)l * DD, fdst);
    }

    // mean pooling over batch
    k_fill<<<cdiv(GG * DD, 256), 256, 0, stream>>>(gsum, 0.0f, GG * DD);
    k_fill<<<1, 256, 0, stream>>>(gcnt, 0.0f, GG);
    k_count<<<cdiv(NN, 256), 256, 0, stream>>>(batch, gcnt);
    k_gsum<<<cdiv((long long)NN * DD, 256), 256, 0, stream>>>(out, batch, gsum);
    k_gdiv<<<cdiv(GG * DD, 256), 256, 0, stream>>>(gsum, gcnt, out + (size_t)NN * DD);
}